// VarianceAdaptor_26379689132746
// MI455X (gfx1250) — compile-verified
//
#include <hip/hip_runtime.h>
#include <hip/hip_bf16.h>

// ---------------- problem constants (from reference) ----------------
#define BB     16
#define TT     512
#define DD     256
#define KK     3
#define RR     (DD * KK)      // 768 = GEMM reduction length
#define MAXOUT 4096
#define LN_EPS 1e-5f
#define HID_ELEMS ((size_t)BB * MAXOUT * DD)   // 16,777,216 floats

typedef __attribute__((ext_vector_type(2))) float v2f;
typedef __attribute__((ext_vector_type(8))) float v8f;

// ---------------- weight repack: (L,D,D,K) OIW -> W'[l][n][r], r=k*D+i
__global__ void __launch_bounds__(256)
va_repack_w(const float* __restrict__ w, float* __restrict__ wp)
{
    const size_t idx = (size_t)blockIdx.x * 256 + threadIdx.x;  // 2*D*R
    if (idx >= 2ull * DD * RR) return;
    const int r = (int)(idx % RR);
    const int n = (int)((idx / RR) % DD);
    const int l = (int)(idx / ((size_t)DD * RR));
    const int k = r >> 8, i = r & 255;
    wp[idx] = w[(((size_t)l * DD + n) * DD + i) * KK + k];
}

// =====================================================================
// Conv1d (K=3, same-pad) + bias + ReLU as a WMMA f32 GEMM.
// Block = 512 threads (16 waves) = 16 timesteps x all 256 out channels.
// Input patch (18 rows incl. halo) staged to LDS with CDNA5 async
// global->LDS copies (ASYNCcnt). Wave w computes the 16x16 tile for
// channels [16w,16w+16) with V_WMMA_F32_16X16X4_F32.
// Reduction r = k*D + i split into 3 affine segments (fixed tap k) so
// all A/B addresses become immediate-offset b64 loads.
//   A[m][r] = in[t0+m+k-1][i]   (LDS row m+k, contiguous pair in i)
//   B[n][r] = W'[n][r]          (contiguous pair -> global_load_b64)
// =====================================================================
__global__ void __launch_bounds__(512)
va_conv_relu(const float* __restrict__ in,     // (B,T,D)
             const float* __restrict__ wp,     // (D,R) repacked, this layer
             const float* __restrict__ bias,   // (D)   this layer
             float* __restrict__ out)          // (B,T,D)
{
    __shared__ float lds[18 * DD];

    const int b    = blockIdx.x >> 5;          // T/16 = 32 tiles per batch
    const int t0   = (blockIdx.x & 31) << 4;
    const int tid  = threadIdx.x;
    const int lane = tid & 31;
    const int wave = tid >> 5;
    const int n0   = wave << 4;

    // ---- async stage: 18 rows x 64 float4 chunks = 1152 copies -------
    {
        const uint32_t lds_base = (uint32_t)(uintptr_t)&lds[0]; // AS3 offset
        const size_t rowbase = (size_t)b * TT;
        for (int s = tid; s < 18 * (DD / 4); s += 512) {
            const int j = s >> 6;              // patch row
            const int q = s & 63;              // float4 within row
            const int t = t0 + j - 1;
            const uint32_t la = lds_base + (uint32_t)s * 16u;
            if (t >= 0 && t < TT) {
                const uint64_t ga =
                    (uint64_t)(uintptr_t)(in + (rowbase + t) * DD + q * 4);
                asm volatile("global_load_async_to_lds_b128 %0, %1, off"
                             :: "v"(la), "v"(ga) : "memory");
            } else {
                *(float4*)&lds[(size_t)s * 4] =
                    make_float4(0.0f, 0.0f, 0.0f, 0.0f);  // halo rows
            }
        }
        asm volatile("s_wait_asynccnt 0x0" ::: "memory");
    }
    __syncthreads();

    const int half = lane >> 4;                // 0: K-pair {0,1}, 1: {2,3}
    const int lrow = lane & 15;                // M for A-halves / N for B
    const float* __restrict__ wn = wp + (size_t)(n0 + lrow) * RR + 2 * half;

    v8f c = {};
#pragma unroll
    for (int k = 0; k < KK; ++k) {
        const float* __restrict__ pA = &lds[(lrow + k) * DD + 2 * half];
        const float* __restrict__ pB = wn + k * DD;
#pragma unroll 8
        for (int i = 0; i < DD; i += 4) {
            const v2f a  = *(const v2f*)(pA + i);
            const v2f bm = *(const v2f*)(pB + i);
            c = __builtin_amdgcn_wmma_f32_16x16x4_f32(
                    false, a, false, bm, (short)0, c, false, false);
        }
    }

    // C/D layout: VGPR j -> M = j + 8*half, N = lane&15
    const int n = n0 + lrow;
    const float bv = bias[n];
#pragma unroll
    for (int j = 0; j < 8; ++j) {
        const int m = j + 8 * half;
        float v = c[j] + bv;
        v = v > 0.0f ? v : 0.0f;               // ReLU
        out[((size_t)b * TT + (t0 + m)) * DD + n] = v;
    }
}

// ---------------- LayerNorm over D=256, one block per (b,t) ----------
__global__ void __launch_bounds__(256)
va_layernorm(const float* __restrict__ in, float* __restrict__ out,
             const float* __restrict__ g, const float* __restrict__ bta)
{
    const int row = blockIdx.x;
    const int tid = threadIdx.x;
    const float v = in[(size_t)row * DD + tid];

    __shared__ float s1[DD], s2[DD];
    s1[tid] = v; s2[tid] = v * v;
    __syncthreads();
    for (int off = 128; off > 0; off >>= 1) {
        if (tid < off) { s1[tid] += s1[tid + off]; s2[tid] += s2[tid + off]; }
        __syncthreads();
    }
    const float m   = s1[0] * (1.0f / DD);
    const float var = s2[0] * (1.0f / DD) - m * m;
    const float r   = rsqrtf(var + LN_EPS);
    out[(size_t)row * DD + tid] = (v - m) * r * g[tid] + bta[tid];
}

// ---------------- linear (D->1) + MSE partial sums -------------------
__global__ void __launch_bounds__(256)
va_linear_loss(const float* __restrict__ h, const float* __restrict__ lw,
               const float* __restrict__ lb, const int* __restrict__ lab,
               float* __restrict__ acc)
{
    const int row = blockIdx.x * 256 + threadIdx.x;   // b*T + t
    const float4* hr = reinterpret_cast<const float4*>(h + (size_t)row * DD);
    const float4* lw4 = reinterpret_cast<const float4*>(lw);
    float dot = lb[0];
#pragma unroll 8
    for (int i = 0; i < DD / 4; ++i) {
        const float4 a = hr[i], w = lw4[i];
        dot += a.x * w.x + a.y * w.y + a.z * w.z + a.w * w.w;
    }
    const float d = dot - (float)lab[row];
    __shared__ float s[256];
    s[threadIdx.x] = d * d;
    __syncthreads();
    for (int off = 128; off > 0; off >>= 1) {
        if (threadIdx.x < off) s[threadIdx.x] += s[threadIdx.x + off];
        __syncthreads();
    }
    if (threadIdx.x == 0) atomicAdd(acc, s[0]);
}

__global__ void va_zero(float* acc) { *acc = 0.0f; }

__global__ void va_loss_write(const float* acc, float* out)
{
    out[HID_ELEMS + BB] = *acc * (1.0f / (BB * TT));
}

// ---------------- length regulator: inclusive scan per batch ---------
__global__ void __launch_bounds__(512)
va_lr_scan(const int* __restrict__ dur, int* __restrict__ cum,
           float* __restrict__ out)
{
    const int b   = blockIdx.x;
    const int tid = threadIdx.x;
    __shared__ int s[TT];
    int d = dur[b * TT + tid];
    s[tid] = d > 0 ? d : 0;
    __syncthreads();
    for (int off = 1; off < TT; off <<= 1) {
        const int v = (tid >= off) ? s[tid - off] : 0;
        __syncthreads();
        s[tid] += v;
        __syncthreads();
    }
    cum[b * TT + tid] = s[tid];
    if (tid == TT - 1) out[HID_ELEMS + b] = (float)s[TT - 1];  // lengths
}

// ---------------- gather: searchsorted(cum, p, 'right') --------------
// block = 256 threads -> 4 output positions x 64 float4 chunks.
__global__ void __launch_bounds__(256)
va_lr_gather(const float* __restrict__ x, const int* __restrict__ cum,
             float* __restrict__ out)
{
    const int blk = blockIdx.x;                 // B * 1024 blocks
    const int b   = blk >> 10;
    const int p   = ((blk & 1023) << 2) + (threadIdx.x >> 6);
    const int q   = threadIdx.x & 63;
    const int* c  = cum + b * TT;
    const int len = c[TT - 1];

    float4 val = make_float4(0.0f, 0.0f, 0.0f, 0.0f);
    if (p < len) {
        int lo = 0, hi = TT;
        while (lo < hi) {
            const int mid = (lo + hi) >> 1;
            if (c[mid] <= p) lo = mid + 1; else hi = mid;
        }
        const int idx = lo < (TT - 1) ? lo : (TT - 1);
        val = reinterpret_cast<const float4*>(
                  x + ((size_t)b * TT + idx) * DD)[q];
    }
    reinterpret_cast<float4*>(out + ((size_t)b * MAXOUT + p) * DD)[q] = val;
}

// =====================================================================
extern "C" void kernel_launch(void* const* d_in, const int* in_sizes, int n_in,
                              void* d_out, int out_size, void* d_ws, size_t ws_size,
                              hipStream_t stream)
{
    const float* x      = (const float*)d_in[0];   // (B,T,D)
    const int*   mfA    = (const int*)  d_in[1];   // (B,T)
    const float* conv_w = (const float*)d_in[4];   // (L,D,D,K)
    const float* conv_b = (const float*)d_in[5];   // (L,D)
    const float* ln_g   = (const float*)d_in[6];   // (L,D)
    const float* ln_b   = (const float*)d_in[7];   // (L,D)
    const float* lin_w  = (const float*)d_in[8];   // (1,D)
    const float* lin_b  = (const float*)d_in[9];   // (1,)

    float* out = (float*)d_out;
    // scratch lives inside the hiddenSeq region of d_out (fully
    // overwritten by va_lr_gather afterwards): hA, hB ping-pong + W'.
    float* hA = out;                                    // B*T*D floats
    float* hB = out + (size_t)BB * TT * DD;             // B*T*D floats
    float* wpk = out + 2ull * BB * TT * DD;             // 2*D*R floats
    float* acc = (float*)d_ws;                          // loss accumulator
    int*   cum = (int*)((char*)d_ws + 256);             // B*T ints

    va_zero<<<1, 1, 0, stream>>>(acc);
    va_repack_w<<<(2 * DD * RR + 255) / 256, 256, 0, stream>>>(conv_w, wpk);

    const float* hin = x;
    for (int l = 0; l < 2; ++l) {
        va_conv_relu<<<BB * (TT / 16), 512, 0, stream>>>(
            hin, wpk + (size_t)l * DD * RR, conv_b + l * DD, hA);
        va_layernorm<<<BB * TT, 256, 0, stream>>>(
            hA, hB, ln_g + l * DD, ln_b + l * DD);
        hin = hB;
    }
    va_linear_loss<<<BB * TT / 256, 256, 0, stream>>>(hB, lin_w, lin_b, mfA, acc);

    va_lr_scan<<<BB, TT, 0, stream>>>(mfA, cum, out);
    va_lr_gather<<<BB * (MAXOUT / 4), 256, 0, stream>>>(x, cum, out);
    va_loss_write<<<1, 1, 0, stream>>>(acc, out);
}